// ECELoss_19748259627502
// MI455X (gfx1250) — compile-verified
//
#include <hip/hip_runtime.h>
#include <hip/hip_bf16.h>
#include <math.h>

#define NBINS       15
#define NW          8           // waves per block (wave32)
#define BLOCK       (NW * 32)
#define COLS        100
#define LANES_USED  25          // 25 lanes * 4 floats = 100
#define BUF_F       128         // floats per stage buffer (512B, 400 used)
#define MAX_BLOCKS  2048

typedef int v4i __attribute__((vector_size(16)));
typedef __attribute__((address_space(1))) v4i* gptr_v4i;
typedef __attribute__((address_space(3))) v4i* lptr_v4i;

// ---- CDNA5 async global->LDS copy (16B per active lane), ASYNCcnt-tracked ----
__device__ __forceinline__ void async_copy16(const float* g, float* l) {
#if defined(__has_builtin) && __has_builtin(__builtin_amdgcn_global_load_async_to_lds_b128)
  __builtin_amdgcn_global_load_async_to_lds_b128(
      (gptr_v4i)(unsigned long long)g,
      (lptr_v4i)(unsigned)(unsigned long long)l,
      0, 0);
#else
  unsigned l32 = (unsigned)(unsigned long long)l;  // flat LDS addr low 32 bits = LDS byte offset
  asm volatile("global_load_async_to_lds_b128 %0, %1, off" :: "v"(l32), "v"(g) : "memory");
#endif
}

#define WAIT_ASYNC(n) asm volatile("s_wait_asynccnt %0" :: "n"(n) : "memory")

__global__ __launch_bounds__(BLOCK) void ece_main(const float* __restrict__ probs,
                                                  const int*   __restrict__ labels,
                                                  float*       __restrict__ partials,
                                                  int nrows) {
  __shared__ float s_buf[NW][2][BUF_F];        // double-buffered row staging per wave
  __shared__ float s_bins[NW][2 * NBINS];      // per-wave [conf sums | corr sums]

  const int tid  = threadIdx.x;
  const int wid  = tid >> 5;
  const int lane = tid & 31;

  if (tid < NW * 2 * NBINS) ((float*)s_bins)[tid] = 0.0f;
  __syncthreads();

  const int wstride = gridDim.x * NW;
  int row = blockIdx.x * NW + wid;

  float* myC = &s_bins[wid][0];
  float* myK = &s_bins[wid][NBINS];

  // prefetch first row for this wave
  if (row < nrows && lane < LANES_USED)
    async_copy16(probs + (size_t)row * COLS + lane * 4, &s_buf[wid][0][lane * 4]);

  int p = 0;
  for (; row < nrows; row += wstride) {
    const int nxt = row + wstride;
    if (nxt < nrows) {                          // wave-uniform
      if (lane < LANES_USED)
        async_copy16(probs + (size_t)nxt * COLS + lane * 4, &s_buf[wid][p ^ 1][lane * 4]);
      WAIT_ASYNC(1);                            // older (current) buffer complete, next in flight
    } else {
      WAIT_ASYNC(0);
    }

    // read current buffer: 4 floats per lane (lanes 0..24)
    float v0 = 0.f, v1 = 0.f, v2 = 0.f, v3 = 0.f;
    float m  = -INFINITY;
    int   mi = 0x7FFFFFFF;
    if (lane < LANES_USED) {
      const float4 v = *(const float4*)&s_buf[wid][p][lane * 4];
      v0 = v.x; v1 = v.y; v2 = v.z; v3 = v.w;
      const int base = lane * 4;
      m = v0; mi = base;                               // strict > keeps first occurrence
      if (v1 > m) { m = v1; mi = base + 1; }
      if (v2 > m) { m = v2; mi = base + 2; }
      if (v3 > m) { m = v3; mi = base + 3; }
    }

    // wave32 butterfly all-reduce: max value, ties -> lowest index
#pragma unroll
    for (int off = 16; off > 0; off >>= 1) {
      const float om = __shfl_xor(m, off, 32);
      const int   oi = __shfl_xor(mi, off, 32);
      if (om > m || (om == m && oi < mi)) { m = om; mi = oi; }
    }

    // sum of exp(x - max); confidence = 1 / sum
    float s = 0.f;
    if (lane < LANES_USED)
      s = __expf(v0 - m) + __expf(v1 - m) + __expf(v2 - m) + __expf(v3 - m);
#pragma unroll
    for (int off = 16; off > 0; off >>= 1) s += __shfl_xor(s, off, 32);

    const float conf = 1.0f / s;
    const int   lbl  = labels[row];
    const float corr = (mi == lbl) ? 1.0f : 0.0f;

    // searchsorted(boundaries, conf, 'left') - 1, clamped; b[0]=0 < conf always
    int cnt = 0;
#pragma unroll
    for (int i = 1; i <= NBINS; ++i)
      cnt += ((float)i * (1.0f / 15.0f) < conf) ? 1 : 0;
    const int bin = cnt > (NBINS - 1) ? (NBINS - 1) : cnt;

    if (lane == 0) {                     // per-wave private region: deterministic, no atomics
      myC[bin] += conf;
      myK[bin] += corr;
    }
    p ^= 1;
  }

  __syncthreads();
  if (tid < 2 * NBINS) {                 // fixed-order cross-wave reduction -> block partials
    float acc = 0.f;
#pragma unroll
    for (int w = 0; w < NW; ++w) acc += s_bins[w][tid];
    partials[(size_t)blockIdx.x * (2 * NBINS) + tid] = acc;
  }
}

__global__ __launch_bounds__(64) void ece_final(const float* __restrict__ partials,
                                                int nblocks,
                                                float* __restrict__ out,
                                                float inv_n) {
  __shared__ float acc[2 * NBINS];
  const int tid = threadIdx.x;
  if (tid < 2 * NBINS) {
    float s = 0.f;
    for (int b = 0; b < nblocks; ++b) s += partials[(size_t)b * (2 * NBINS) + tid];
    acc[tid] = s;
  }
  __syncthreads();
  if (tid == 0) {
    float e = 0.f;
#pragma unroll
    for (int j = 0; j < NBINS; ++j) e += fabsf(acc[j] - acc[NBINS + j]);
    out[0] = e * inv_n;
  }
}

extern "C" void kernel_launch(void* const* d_in, const int* in_sizes, int n_in,
                              void* d_out, int out_size, void* d_ws, size_t ws_size,
                              hipStream_t stream) {
  const float* probs  = (const float*)d_in[0];
  const int*   labels = (const int*)d_in[1];
  const int    nrows  = in_sizes[1];          // labels count == N

  int B = MAX_BLOCKS;
  const size_t per_block = (size_t)(2 * NBINS) * sizeof(float);
  if (ws_size < (size_t)B * per_block) {
    B = (int)(ws_size / per_block);
    if (B < 1) B = 1;
  }

  ece_main<<<B, BLOCK, 0, stream>>>(probs, labels, (float*)d_ws, nrows);
  ece_final<<<1, 64, 0, stream>>>((const float*)d_ws, B, (float*)d_out,
                                  1.0f / (float)nrows);
}